// TargetAssigner_45784351375629
// MI455X (gfx1250) — compile-verified
//
#include <hip/hip_runtime.h>
#include <math.h>

// Problem constants (from reference)
#define BB 4
#define NN 256
#define HH 400
#define WW 400
#define HWSZ (HH * WW)
#define NCH 15          // 4 heatmap + 11 reg channels
#define TILE 32
#define LOG2E 1.4426950408889634f
// f32 exp underflows to exactly 0 for d2/(2*sigma^2) > ~104; 105 is a safe cutoff.
#define CUT_MULT 105.0f

// ---------------------------------------------------------------------------
// Kernel A: per-tile heatmap max + zero-fill of reg channels.
// CDNA5 path: async global->LDS DMA of the box table, s_wait_asynccnt fence.
// ---------------------------------------------------------------------------
__global__ __launch_bounds__(256)
void ta_heatmap_kernel(const float* __restrict__ gt, float* __restrict__ out)
{
    __shared__ float    s_raw[NN * 10];            // staged boxes (10,240 B)
    __shared__ float    s_cx[NN], s_cy[NN];        // box grid centers (as float)
    __shared__ float    s_k[NN];                   // -log2e / (2*sigma^2)
    __shared__ float    s_cut[NN];                 // d2 cutoff (underflow radius^2)
    __shared__ int      s_cls[NN];
    __shared__ unsigned s_cnt;

    const int tid  = threadIdx.x;
    const int tx0  = blockIdx.x * TILE;
    const int ty0  = blockIdx.y * TILE;
    const int b    = blockIdx.z;

    if (tid == 0) s_cnt = 0u;

    // ---- stage this batch's 256x10 f32 box table into LDS asynchronously ----
    {
        const char*    gbase = (const char*)(gt + (size_t)b * NN * 10);
        const uint32_t lds0  = (uint32_t)(size_t)(&s_raw[0]);
        const int totalChunks = (NN * 10 * 4) / 16;   // 640 x 16B
        for (int c = tid; c < totalChunks; c += 256) {
            uint32_t    loff = lds0 + (uint32_t)c * 16u;
            const char* ga   = gbase + (size_t)c * 16u;
            // gfx1250 async DMA: LDS byte address in vdst VGPR, 64-bit global
            // address in VGPR pair, tracked by ASYNCcnt (not LOADcnt).
            asm volatile("global_load_async_to_lds_b128 %0, %1, off"
                         :: "v"(loff), "v"(ga)
                         : "memory");
        }
        asm volatile("s_wait_asynccnt 0" ::: "memory");
    }
    __syncthreads();   // publish LDS data + s_cnt init across the workgroup

    // ---- per-box derived params + tile culling (one box per thread) ----
    {
        const float* bx = &s_raw[tid * 10];
        float x = bx[0], y = bx[1], z = bx[2], w = bx[3], l = bx[4];
        bool nonzero = (fabsf(x) + fabsf(y) + fabsf(z)) > 0.0f;
        float gx = (x - 0.0f)   / 0.1f;   // (x - PC_MIN_X) / VOXEL
        float gy = (y + 39.68f) / 0.1f;   // (y - PC_MIN_Y) / VOXEL
        int gxi = (int)floorf(gx);
        int gyi = (int)floorf(gy);
        bool valid = nonzero && gxi >= 0 && gxi < WW && gyi >= 0 && gyi < HH;
        if (valid) {
            float sig    = fmaxf(sqrtf(w * w + l * l) * 0.5f, 2.0f) * (1.0f / 3.0f);
            float two_s2 = 2.0f * sig * sig;          // <= 1.78
            float cut    = CUT_MULT * two_s2;          // radius^2 <= ~187
            float cx = (float)gxi, cy = (float)gyi;
            // distance from box center to this tile's rectangle
            float tx1 = (float)min(tx0 + TILE - 1, WW - 1);
            float ty1 = (float)min(ty0 + TILE - 1, HH - 1);
            float dx = cx - fminf(fmaxf(cx, (float)tx0), tx1);
            float dy = cy - fminf(fmaxf(cy, (float)ty0), ty1);
            if (dx * dx + dy * dy <= cut) {
                unsigned idx = atomicAdd(&s_cnt, 1u);  // ds_add_rtn_u32 compaction
                s_cx[idx]  = cx;
                s_cy[idx]  = cy;
                s_k[idx]   = -LOG2E / two_s2;
                s_cut[idx] = cut;
                s_cls[idx] = (int)bx[7];
            }
        }
    }
    __syncthreads();

    // ---- each thread owns 4 pixels (same column, rows 8 apart) ----
    const int lx    = tid & 31;
    const int lySub = tid >> 5;
    const int px    = tx0 + lx;
    const float pxf = (float)px;

    float m[4][4];                 // [row][class] running maxima
    #pragma unroll
    for (int r = 0; r < 4; ++r)
        #pragma unroll
        for (int c = 0; c < 4; ++c) m[r][c] = 0.0f;

    float pyf[4];
    #pragma unroll
    for (int r = 0; r < 4; ++r) pyf[r] = (float)(ty0 + lySub + r * 8);

    const unsigned cnt = s_cnt;
    for (unsigned j = 0; j < cnt; ++j) {
        float cx = s_cx[j], cy = s_cy[j], k = s_k[j], cut = s_cut[j];
        int   cls = s_cls[j];
        bool c0 = (cls == 0), c1 = (cls == 1), c2 = (cls == 2), c3 = (cls == 3);
        float dx  = pxf - cx;
        float dx2 = dx * dx;
        #pragma unroll
        for (int r = 0; r < 4; ++r) {
            float dy = pyf[r] - cy;
            float d2 = fmaf(dy, dy, dx2);
            // exp(-d2/(2s^2)) == exp2(d2 * -log2e/(2s^2)); 0 past underflow radius
            float e = (d2 <= cut) ? exp2f(d2 * k) : 0.0f;
            m[r][0] = fmaxf(m[r][0], c0 ? e : 0.0f);
            m[r][1] = fmaxf(m[r][1], c1 ? e : 0.0f);
            m[r][2] = fmaxf(m[r][2], c2 ? e : 0.0f);
            m[r][3] = fmaxf(m[r][3], c3 ? e : 0.0f);
        }
    }

    // ---- write 4 heatmap channels + zero the 11 reg channels (store-bound) ----
    if (px < WW) {
        float* ob = out + (size_t)b * NCH * HWSZ;
        #pragma unroll
        for (int r = 0; r < 4; ++r) {
            int py = ty0 + lySub + r * 8;
            if (py < HH) {
                float* p = ob + (size_t)py * WW + px;
                #pragma unroll
                for (int c = 0; c < 4; ++c) p[(size_t)c * HWSZ] = m[r][c];
                #pragma unroll
                for (int k2 = 4; k2 < NCH; ++k2) p[(size_t)k2 * HWSZ] = 0.0f;
            }
        }
    }
}

// ---------------------------------------------------------------------------
// Kernel B: sparse scatter of the 11 regression channels (1024 boxes total).
// Runs after kernel A on the same stream (stream order = reg zeros first).
// ---------------------------------------------------------------------------
__global__ __launch_bounds__(256)
void ta_scatter_kernel(const float* __restrict__ gt, float* __restrict__ out)
{
    int i = blockIdx.x * blockDim.x + threadIdx.x;   // 0 .. B*N-1
    if (i >= BB * NN) return;
    int b = i / NN;

    const float* bx = gt + (size_t)i * 10;
    float x = bx[0], y = bx[1], z = bx[2];
    float w = bx[3], l = bx[4], h = bx[5], yaw = bx[6];
    float vx = bx[8], vy = bx[9];

    bool nonzero = (fabsf(x) + fabsf(y) + fabsf(z)) > 0.0f;
    float gx = (x - 0.0f)   / 0.1f;
    float gy = (y + 39.68f) / 0.1f;
    int gxi = (int)floorf(gx);
    int gyi = (int)floorf(gy);
    if (!(nonzero && gxi >= 0 && gxi < WW && gyi >= 0 && gyi < HH)) return;

    float xoff = gx - (float)gxi;
    float yoff = gy - (float)gyi;

    float vals[11] = { xoff, yoff, z, w, l, h,
                       sinf(yaw), cosf(yaw), vx, vy, 1.0f };

    float* p = out + (size_t)b * NCH * HWSZ + (size_t)4 * HWSZ
                   + (size_t)gyi * WW + gxi;
    #pragma unroll
    for (int k = 0; k < 11; ++k) p[(size_t)k * HWSZ] = vals[k];
}

// ---------------------------------------------------------------------------
extern "C" void kernel_launch(void* const* d_in, const int* in_sizes, int n_in,
                              void* d_out, int out_size, void* d_ws, size_t ws_size,
                              hipStream_t stream)
{
    (void)in_sizes; (void)n_in; (void)out_size; (void)d_ws; (void)ws_size;
    const float* gt  = (const float*)d_in[0];   // (B, N, 10) f32
    // d_in[1] = spatial_features: shape-only in the reference, unused.
    float* out = (float*)d_out;                 // (B, 15, 400, 400) f32

    dim3 gridA((WW + TILE - 1) / TILE, (HH + TILE - 1) / TILE, BB);  // 13x13x4
    ta_heatmap_kernel<<<gridA, 256, 0, stream>>>(gt, out);

    dim3 gridB((BB * NN + 255) / 256);                                // 4 blocks
    ta_scatter_kernel<<<gridB, 256, 0, stream>>>(gt, out);
}